// Encoder_35811437314561
// MI455X (gfx1250) — compile-verified
//
#include <hip/hip_runtime.h>
#include <hip/hip_bf16.h>
#include <math.h>

// ---------------------------------------------------------------------------
// Problem constants (from the reference)
// ---------------------------------------------------------------------------
#define NN   10000
#define DD   32      // neighbors
#define HIDN 256
#define NH   8       // heads
#define DH   32      // head dim
#define KK   8       // top-k
#define LL   2       // layers
#define FF   2       // ffn layers
#define EPS_LN 1e-5f

typedef float v2f __attribute__((ext_vector_type(2)));
typedef float v8f __attribute__((ext_vector_type(8)));

__device__ __forceinline__ float mish_f(float x) {
    // x * tanh(softplus(x)), stable for large x
    float sp = (x > 20.0f) ? x : log1pf(expf(x));
    return x * tanhf(sp);
}

// ---------------------------------------------------------------------------
// GEMM: out[n][o] = act( sum_k A[n][k] * W[o][k] + bias[o] )
// One wave computes a 16x16 tile via V_WMMA_F32_16X16X4_F32.
// Block = 128 threads (4 waves). Waves share a 16x256 A tile in LDS.
// grid = (nrows/16, HID/64)
// ---------------------------------------------------------------------------
#define AS_STRIDE 258  // pad 256 -> 258 words to avoid LDS bank conflicts

template <int ACT>
__global__ __launch_bounds__(128)
void gemm_bias_act(const float* __restrict__ A,
                   const float* __restrict__ W,
                   const float* __restrict__ bias,
                   float* __restrict__ out)
{
    __shared__ float As[16 * AS_STRIDE];

    const int m0   = blockIdx.x * 16;
    const int wave = threadIdx.x >> 5;
    const int lane = threadIdx.x & 31;
    const int n0   = blockIdx.y * 64 + wave * 16;

    // Cooperative stage of A tile (16 rows x 256 cols), coalesced along k
    for (int idx = threadIdx.x; idx < 16 * HIDN; idx += 128) {
        int r = idx >> 8;      // row within tile
        int k = idx & 255;     // column
        As[r * AS_STRIDE + k] = A[(size_t)(m0 + r) * HIDN + k];
    }
    __syncthreads();

    const int half = lane >> 4;   // 0: lanes 0-15, 1: lanes 16-31
    const int r    = lane & 15;

    v8f c = {0.f, 0.f, 0.f, 0.f, 0.f, 0.f, 0.f, 0.f};

    const float* wrow = W + (size_t)(n0 + r) * HIDN;   // B[k][n] = W[n][k]
    const float* arow = As + r * AS_STRIDE;

    // K loop: 256 in steps of 4
    for (int k0 = 0; k0 < HIDN; k0 += 4) {
        // A frag: v0 = A[r][k0+2h], v1 = A[r][k0+2h+1]   (ISA 16x4 f32 layout)
        v2f a = *(const v2f*)(arow + k0 + 2 * half);
        // B frag: v0 = B[k0+2h][n0+r], v1 = B[k0+2h+1][n0+r]
        v2f b = *(const v2f*)(wrow + k0 + 2 * half);
        c = __builtin_amdgcn_wmma_f32_16x16x4_f32(
                false, a, false, b, (short)0, c, false, false);
    }

    // D layout: c[v] = out[m0 + v + 8*half][n0 + r]
    const float bv = bias[n0 + r];
#pragma unroll
    for (int v = 0; v < 8; ++v) {
        float val = c[v] + bv;
        if (ACT) val = mish_f(val);
        out[(size_t)(m0 + v + 8 * half) * HIDN + (n0 + r)] = val;
    }
}

// ---------------------------------------------------------------------------
// LayerNorm (optionally fused residual): out = (base ? base : 0) + LN(x)
// One wave (32 lanes) per row; 8 rows per 256-thread block.
// ---------------------------------------------------------------------------
__global__ __launch_bounds__(256)
void layernorm_k(const float* __restrict__ x,
                 const float* __restrict__ base,   // may be nullptr
                 const float* __restrict__ gamma,
                 const float* __restrict__ beta,
                 float* __restrict__ out, int nrows)
{
    const int row  = blockIdx.x * (blockDim.x >> 5) + (threadIdx.x >> 5);
    const int lane = threadIdx.x & 31;
    if (row >= nrows) return;

    const float* xr = x + (size_t)row * HIDN;
    float vals[8];
    float s = 0.f, ss = 0.f;
#pragma unroll
    for (int j = 0; j < 8; ++j) {
        float v = xr[lane + 32 * j];
        vals[j] = v;
        s  += v;
        ss += v * v;
    }
#pragma unroll
    for (int o = 16; o > 0; o >>= 1) {
        s  += __shfl_xor(s,  o, 32);
        ss += __shfl_xor(ss, o, 32);
    }
    const float mu  = s * (1.0f / HIDN);
    const float var = ss * (1.0f / HIDN) - mu * mu;
    const float inv = rsqrtf(var + EPS_LN);

    float* orow = out + (size_t)row * HIDN;
    const float* brow = base ? (base + (size_t)row * HIDN) : nullptr;
#pragma unroll
    for (int j = 0; j < 8; ++j) {
        int ci = lane + 32 * j;
        float v = (vals[j] - mu) * inv * gamma[ci] + beta[ci];
        if (brow) v += brow[ci];
        orow[ci] = v;
    }
}

// ---------------------------------------------------------------------------
// Attention: one block (256 thr = 8 waves) per node; wave = head.
//   wave 0: stable top-K rank of edge weights via lane shuffles, normalize.
//   each wave: per-neighbor score = q.k over DH, in-wave softmax, then each
//   lane owns one output channel and accumulates over the 32 neighbors.
// ---------------------------------------------------------------------------
__global__ __launch_bounds__(256)
void attn_k(const float* __restrict__ q,
            const float* __restrict__ k,
            const float* __restrict__ v,
            const int*   __restrict__ neigh,
            const float* __restrict__ edge_w,
            float* __restrict__ out)
{
    const int n    = blockIdx.x;
    const int lane = threadIdx.x & 31;
    const int wv   = threadIdx.x >> 5;

    __shared__ int   s_idx[DD];
    __shared__ float s_wn[DD];

    if (wv == 0) {
        float w  = edge_w[(size_t)n * DD + lane];
        int   id = neigh[(size_t)n * DD + lane];
        s_idx[lane] = id;
        // rank = #{j : w[j] > w[i]  or  (w[j]==w[i] and j<i)}  (stable argsort of -w)
        int rank = 0;
#pragma unroll
        for (int j = 0; j < DD; ++j) {
            float wj = __shfl(w, j, 32);
            rank += (wj > w) || (wj == w && j < lane);
        }
        float wm = (rank < KK) ? w : 0.0f;
        float ssum = wm;
#pragma unroll
        for (int o = 16; o > 0; o >>= 1) ssum += __shfl_xor(ssum, o, 32);
        s_wn[lane] = wm / (ssum + 1e-5f);
    }
    __syncthreads();

    const int hd   = wv;          // head
    const int d    = lane;        // neighbor slot
    const int base = hd * DH;
    const int src  = s_idx[d];

    const float* qrow = q + (size_t)n   * HIDN + base;
    const float* krow = k + (size_t)src * HIDN + base;
    float score = 0.f;
#pragma unroll
    for (int c = 0; c < DH; ++c) score += qrow[c] * krow[c];

    float s = score * s_wn[d] * 0.1767766952966369f;   // 1/sqrt(DH)

    // softmax across the 32 lanes (= neighbor axis)
    float m = s;
#pragma unroll
    for (int o = 16; o > 0; o >>= 1) m = fmaxf(m, __shfl_xor(m, o, 32));
    float e = expf(s - m);
    float tot = e;
#pragma unroll
    for (int o = 16; o > 0; o >>= 1) tot += __shfl_xor(tot, o, 32);
    float attn = e / tot;

    // lane = output channel c; accumulate over neighbors dd
    float acc = 0.f;
#pragma unroll 4
    for (int dd = 0; dd < DD; ++dd) {
        float a = __shfl(attn, dd, 32);
        acc += a * v[(size_t)s_idx[dd] * HIDN + base + lane];
    }
    out[(size_t)n * HIDN + base + lane] = acc;
}

// ---------------------------------------------------------------------------
// Driver
// ---------------------------------------------------------------------------
extern "C" void kernel_launch(void* const* d_in, const int* in_sizes, int n_in,
                              void* d_out, int out_size, void* d_ws, size_t ws_size,
                              hipStream_t stream) {
    const float* h_in   = (const float*)d_in[0];
    const int*   neigh  = (const int*)  d_in[1];
    const float* edge_w = (const float*)d_in[2];
    const float* Wq     = (const float*)d_in[3];
    const float* bq     = (const float*)d_in[4];
    const float* Wk     = (const float*)d_in[5];
    const float* bk     = (const float*)d_in[6];
    const float* Wv     = (const float*)d_in[7];
    const float* bv     = (const float*)d_in[8];
    const float* Wo     = (const float*)d_in[9];
    const float* bo     = (const float*)d_in[10];
    const float* Wf     = (const float*)d_in[11];  // [L,F,HID,HID]
    const float* bf     = (const float*)d_in[12];  // [L,F,HID]
    const float* gamma  = (const float*)d_in[13];
    const float* beta   = (const float*)d_in[14];

    const size_t BUF = (size_t)NN * HIDN;
    float* ws   = (float*)d_ws;
    float* hcur = ws + 0 * BUF;   // running hidden state
    float* qb   = ws + 1 * BUF;
    float* kb   = ws + 2 * BUF;
    float* vb   = ws + 3 * BUF;
    float* ab   = ws + 4 * BUF;   // attention output
    float* tb   = ws + 5 * BUF;   // temp

    const dim3 gGemm(NN / 16, HIDN / 64);
    const dim3 bGemm(128);
    const dim3 gLN((NN + 7) / 8);
    const dim3 bLN(256);

    // h0 = LN(h_in)
    layernorm_k<<<gLN, bLN, 0, stream>>>(h_in, nullptr, gamma, beta, hcur, NN);

    for (int i = 0; i < LL; ++i) {
        const size_t wOff = (size_t)i * HIDN * HIDN;
        const size_t bOff = (size_t)i * HIDN;

        // Q/K/V projections (bias only)
        gemm_bias_act<0><<<gGemm, bGemm, 0, stream>>>(hcur, Wq + wOff, bq + bOff, qb);
        gemm_bias_act<0><<<gGemm, bGemm, 0, stream>>>(hcur, Wk + wOff, bk + bOff, kb);
        gemm_bias_act<0><<<gGemm, bGemm, 0, stream>>>(hcur, Wv + wOff, bv + bOff, vb);

        // neighbor attention with top-K edge gating
        attn_k<<<NN, 256, 0, stream>>>(qb, kb, vb, neigh, edge_w, ab);

        // O projection + mish
        gemm_bias_act<1><<<gGemm, bGemm, 0, stream>>>(ab, Wo + wOff, bo + bOff, tb);

        // h1 = h0 + LN(mhsa)   -> store in qb
        layernorm_k<<<gLN, bLN, 0, stream>>>(tb, hcur, gamma, beta, qb, NN);

        // FFN: two mish GEMMs
        const float* Wf0 = Wf + ((size_t)(i * FF + 0) * HIDN * HIDN);
        const float* Wf1 = Wf + ((size_t)(i * FF + 1) * HIDN * HIDN);
        const float* bf0 = bf + ((size_t)(i * FF + 0) * HIDN);
        const float* bf1 = bf + ((size_t)(i * FF + 1) * HIDN);
        gemm_bias_act<1><<<gGemm, bGemm, 0, stream>>>(qb, Wf0, bf0, kb);
        gemm_bias_act<1><<<gGemm, bGemm, 0, stream>>>(kb, Wf1, bf1, vb);

        // h2 = h1 + LN(ffn) -> back into hcur
        layernorm_k<<<gLN, bLN, 0, stream>>>(vb, qb, gamma, beta, hcur, NN);
    }

    // out = LN(h)
    layernorm_k<<<gLN, bLN, 0, stream>>>(hcur, nullptr, gamma, beta, (float*)d_out, NN);
}